// NodeClassification_42116449305312
// MI455X (gfx1250) — compile-verified
//
#include <hip/hip_runtime.h>

typedef __attribute__((ext_vector_type(2))) float v2f;
typedef __attribute__((ext_vector_type(8))) float v8f;

#define KDIM 128  // input feature dim is 128 for all three layers

// -------------------------------------------------------------- zero fill
__global__ void k_zero(float* __restrict__ p, long long n) {
  long long i = (long long)blockIdx.x * blockDim.x + threadIdx.x;
  long long stride = (long long)gridDim.x * blockDim.x;
  for (; i < n; i += stride) p[i] = 0.0f;
}

// -------------------------------------------------------------- degree accumulation
__global__ void k_degrees(const int* __restrict__ src, const int* __restrict__ dst,
                          float* __restrict__ outdeg, float* __restrict__ indeg, int nE) {
  int e = blockIdx.x * blockDim.x + threadIdx.x;
  if (e < nE) {
    atomicAdd(&outdeg[src[e]], 1.0f);
    atomicAdd(&indeg[dst[e]], 1.0f);
  }
}

// clip(deg,1)^-0.5 in place (runs over outdeg and indeg as one contiguous 2N range)
__global__ void k_isqrt(float* __restrict__ p, int n) {
  int i = blockIdx.x * blockDim.x + threadIdx.x;
  if (i < n) {
    float d = p[i];
    d = d < 1.0f ? 1.0f : d;
    p[i] = rsqrtf(d);
  }
}

// -------------------------------------------------------------- scaled GEMM via WMMA f32
// Out[m, :] = (scale[m] * H[m, :]) @ Wm,  H: n_rows x 128, Wm: 128 x NDIM (row major)
// 256 threads = 8 waves; each wave computes a 16-row slab across all NDIM columns.
// A frag (16x4 f32): lane(0-15)=M, half-wave g selects K pair {2g, 2g+1}.
// C/D (16x16 f32): vgpr r, lanes0-15 -> M=r, lanes16-31 -> M=r+8.
template <int NDIM>
__launch_bounds__(256)
__global__ void k_gemm_scaled(const float* __restrict__ H, const float* __restrict__ Wm,
                              const float* __restrict__ scale, float* __restrict__ Out,
                              int n_rows) {
  __shared__ float sW[KDIM * NDIM];  // 64KB (NDIM=128) / 32KB (NDIM=64)
  const int tid = threadIdx.x;
  for (int i = tid; i < KDIM * NDIM; i += 256) sW[i] = Wm[i];
  __syncthreads();

  const int wave = tid >> 5;
  const int lane = tid & 31;
  const int g    = lane >> 4;   // half-wave: K-pair select
  const int ml   = lane & 15;   // M (A frag) / N (B frag) within tile
  const int row0 = (blockIdx.x * 8 + wave) * 16;
  if (row0 >= n_rows) return;   // whole-wave uniform exit: EXEC stays all-1s for WMMA

  int m = row0 + ml;
  if (m >= n_rows) m = n_rows - 1;       // clamp (garbage rows never stored)
  const float s = scale[m];
  const float* __restrict__ hrow = H + (size_t)m * KDIM;

  constexpr int NT = NDIM / 16;
  v8f acc[NT];
#pragma unroll
  for (int t = 0; t < NT; ++t)
    acc[t] = (v8f){0.f, 0.f, 0.f, 0.f, 0.f, 0.f, 0.f, 0.f};

  for (int k = 0; k < KDIM; k += 4) {
    const int ka = k + 2 * g;
    v2f a;
    a.x = hrow[ka] * s;       // contiguous pair -> b64 load; out_isqrt fused here
    a.y = hrow[ka + 1] * s;
#pragma unroll
    for (int t = 0; t < NT; ++t) {
      v2f b;
      b.x = sW[ka * NDIM + t * 16 + ml];
      b.y = sW[(ka + 1) * NDIM + t * 16 + ml];
      acc[t] = __builtin_amdgcn_wmma_f32_16x16x4_f32(
          /*neg_a=*/false, a, /*neg_b=*/false, b,
          /*c_mod=*/(short)0, acc[t], /*reuse_a=*/false, /*reuse_b=*/false);
    }
  }

#pragma unroll
  for (int t = 0; t < NT; ++t) {
#pragma unroll
    for (int r = 0; r < 8; ++r) {
      const int mm = row0 + 8 * g + r;
      if (mm < n_rows) Out[(size_t)mm * NDIM + t * 16 + ml] = acc[t][r];
    }
  }
}

// -------------------------------------------------------------- edge scatter-add (SpMM)
// FDIM/4 consecutive threads per edge: float4 gather from src row, 4 f32 atomics to dst row.
template <int FDIM>
__global__ void k_scatter(const int* __restrict__ src, const int* __restrict__ dst,
                          const float* __restrict__ X, float* __restrict__ Y, int nE) {
  constexpr int TPE = FDIM / 4;  // 32 (128-wide) or 16 (64-wide)
  long long idx = (long long)blockIdx.x * blockDim.x + threadIdx.x;
  long long e = idx / TPE;
  if (e >= nE) return;
  const int f = (int)(idx - e * TPE) * 4;
  const int sN = src[e];
  const int dN = dst[e];
  const float4 v = *reinterpret_cast<const float4*>(X + (size_t)sN * FDIM + f);
  float* yp = Y + (size_t)dN * FDIM + f;
  atomicAdd(yp + 0, v.x);
  atomicAdd(yp + 1, v.y);
  atomicAdd(yp + 2, v.z);
  atomicAdd(yp + 3, v.w);
}

// -------------------------------------------------------------- in_isqrt scale + ReLU
__global__ void k_epilogue(const float* __restrict__ A, const float* __restrict__ isq,
                           float* __restrict__ O, long long n4, int q /* = dim/4 */) {
  long long i = (long long)blockIdx.x * blockDim.x + threadIdx.x;
  if (i >= n4) return;
  long long row = i / q;
  const float s = isq[row];
  float4 v = reinterpret_cast<const float4*>(A)[i];
  float4 o;
  o.x = fmaxf(v.x * s, 0.f);
  o.y = fmaxf(v.y * s, 0.f);
  o.z = fmaxf(v.z * s, 0.f);
  o.w = fmaxf(v.w * s, 0.f);
  reinterpret_cast<float4*>(O)[i] = o;
}

// -------------------------------------------------------------- driver
extern "C" void kernel_launch(void* const* d_in, const int* in_sizes, int n_in,
                              void* d_out, int out_size, void* d_ws, size_t ws_size,
                              hipStream_t stream) {
  const float* h  = (const float*)d_in[0];
  const int*   ei = (const int*)d_in[1];
  const float* W1 = (const float*)d_in[2];
  const float* W2 = (const float*)d_in[3];
  const float* W3 = (const float*)d_in[4];
  float* out = (float*)d_out;

  const int N = in_sizes[0] / 128;  // 50000
  const int E = in_sizes[1] / 2;    // 1.6M
  const int* src = ei;
  const int* dst = ei + E;

  // workspace layout (all 16B aligned): buf0 | buf1 | outisq | inisq
  float* buf0   = (float*)d_ws;                 // N*128
  float* buf1   = buf0 + (size_t)N * 128;       // N*128
  float* outisq = buf1 + (size_t)N * 128;       // N
  float* inisq  = outisq + N;                   // N  (contiguous with outisq)

  const int TB = 256;
  const long long nfeat   = (long long)N * 128;
  const long long nfeat64 = (long long)N * 64;
  const long long sc128   = (long long)E * 32;
  const long long sc64    = (long long)E * 16;
  const int gemmBlocks = (((N + 15) / 16) + 7) / 8;  // 8 wave-slabs of 16 rows per block

  // normalization coefficients
  k_zero<<<512, TB, 0, stream>>>(outisq, (long long)2 * N);
  k_degrees<<<(E + TB - 1) / TB, TB, 0, stream>>>(src, dst, outisq, inisq, E);
  k_isqrt<<<(2 * N + TB - 1) / TB, TB, 0, stream>>>(outisq, 2 * N);

  // ---- Layer 1: h -> buf0 -> (scatter) buf1 -> (epilogue) buf0
  k_gemm_scaled<128><<<gemmBlocks, TB, 0, stream>>>(h, W1, outisq, buf0, N);
  k_zero<<<2048, TB, 0, stream>>>(buf1, nfeat);
  k_scatter<128><<<(int)((sc128 + TB - 1) / TB), TB, 0, stream>>>(src, dst, buf0, buf1, E);
  k_epilogue<<<(int)((nfeat / 4 + TB - 1) / TB), TB, 0, stream>>>(buf1, inisq, buf0, nfeat / 4, 32);

  // ---- Layer 2: buf0 -> buf1 -> (scatter) buf0 -> (epilogue) buf1
  k_gemm_scaled<128><<<gemmBlocks, TB, 0, stream>>>(buf0, W2, outisq, buf1, N);
  k_zero<<<2048, TB, 0, stream>>>(buf0, nfeat);
  k_scatter<128><<<(int)((sc128 + TB - 1) / TB), TB, 0, stream>>>(src, dst, buf1, buf0, E);
  k_epilogue<<<(int)((nfeat / 4 + TB - 1) / TB), TB, 0, stream>>>(buf0, inisq, buf1, nfeat / 4, 32);

  // ---- Layer 3 (64-wide): buf1 -> buf0 -> (scatter) buf1 -> (epilogue) d_out
  k_gemm_scaled<64><<<gemmBlocks, TB, 0, stream>>>(buf1, W3, outisq, buf0, N);
  k_zero<<<1024, TB, 0, stream>>>(buf1, nfeat64);  // buf1 head reused as agg
  k_scatter<64><<<(int)((sc64 + TB - 1) / TB), TB, 0, stream>>>(src, dst, buf0, buf1, E);
  k_epilogue<<<(int)((nfeat64 / 4 + TB - 1) / TB), TB, 0, stream>>>(buf1, inisq, out, nfeat64 / 4, 16);
}